// GraphAutoencoder_23871428231489
// MI455X (gfx1250) — compile-verified
//
#include <hip/hip_runtime.h>

#define N_NODES 12288
#define N_EDGES 393216

typedef float v2f __attribute__((ext_vector_type(2)));
typedef float v8f __attribute__((ext_vector_type(8)));

// ---------------- degree / norm precompute ----------------

__global__ void k_init_deg(float* __restrict__ deg) {
    int i = blockIdx.x * blockDim.x + threadIdx.x;
    deg[i] = 1.0f;                       // self-loop included
}

__global__ void k_count_deg(const int* __restrict__ dst, float* __restrict__ deg) {
    int e = blockIdx.x * blockDim.x + threadIdx.x;
    atomicAdd(&deg[dst[e]], 1.0f);
}

__global__ void k_finish_deg(const float* __restrict__ deg,
                             float* __restrict__ dinvs, float* __restrict__ deginv) {
    int i = blockIdx.x * blockDim.x + threadIdx.x;
    float d = deg[i];
    dinvs[i]  = rsqrtf(d);
    deginv[i] = 1.0f / d;
}

__global__ void k_norm(const int* __restrict__ src, const int* __restrict__ dst,
                       const float* __restrict__ dinvs, float* __restrict__ norm) {
    int e = blockIdx.x * blockDim.x + threadIdx.x;
    norm[e] = dinvs[src[e]] * dinvs[dst[e]];
}

// ---------------- fp32 WMMA GEMM: H[N,Fout] = X[N,K] @ W[K,Fout] ----------------
// One wave computes one 16x16 output tile using V_WMMA_F32_16X16X4_F32.
// A layout (16x4 f32): lane L holds M=L&15, K = 2*(L>>4) + {0,1} in {a.x,a.y}
// B layout (4x16 f32): lane L holds N=L&15, K = 2*(L>>4) + {0,1} in {b.x,b.y}
// C layout: vgpr r, lane L -> row = 8*(L>>4) + r, col = L&15

__global__ void k_gemm_xw(const float* __restrict__ X, const float* __restrict__ W,
                          float* __restrict__ H, int K, int Fout) {
    int gwave = (blockIdx.x * blockDim.x + threadIdx.x) >> 5;
    int lane  = threadIdx.x & 31;
    int tiles_n = Fout >> 4;
    int tile_m = gwave / tiles_n;
    int tile_n = gwave - tile_m * tiles_n;
    int m0 = tile_m << 4, n0 = tile_n << 4;
    int mm = lane & 15;
    int kk = (lane >> 4) << 1;           // 0 or 2

    const float* arow = X + (size_t)(m0 + mm) * K;
    v8f c = {0.f, 0.f, 0.f, 0.f, 0.f, 0.f, 0.f, 0.f};
    for (int k0 = 0; k0 < K; k0 += 4) {
        v2f a, b;
        a.x = arow[k0 + kk];
        a.y = arow[k0 + kk + 1];
        b.x = W[(size_t)(k0 + kk)     * Fout + n0 + mm];
        b.y = W[(size_t)(k0 + kk + 1) * Fout + n0 + mm];
        c = __builtin_amdgcn_wmma_f32_16x16x4_f32(false, a, false, b,
                                                  (short)0, c, false, false);
    }
    int rbase = m0 + ((lane >> 4) << 3);
#pragma unroll
    for (int r = 0; r < 8; ++r)
        H[(size_t)(rbase + r) * Fout + n0 + mm] = c[r];
}

// ---------------- a_hat = S @ S^T  (S: [N,64]) ----------------
// grid (N/16/8, N/16), block 256 (8 waves); wave w -> col tile blockIdx.x*8+w

__global__ void k_gemm_sst(const float* __restrict__ S, float* __restrict__ AH) {
    int wave = threadIdx.x >> 5;
    int lane = threadIdx.x & 31;
    int n0 = (((blockIdx.x << 3) + wave) << 4);
    int m0 = (blockIdx.y << 4);
    int mm = lane & 15;
    int kk = (lane >> 4) << 1;

    const float* arow = S + (size_t)(m0 + mm) * 64;   // A = S rows
    const float* brow = S + (size_t)(n0 + mm) * 64;   // B[k][n] = S[n][k]
    v8f c = {0.f, 0.f, 0.f, 0.f, 0.f, 0.f, 0.f, 0.f};
#pragma unroll
    for (int k0 = 0; k0 < 64; k0 += 4) {
        v2f a, b;
        a.x = arow[k0 + kk];
        a.y = arow[k0 + kk + 1];
        b.x = brow[k0 + kk];
        b.y = brow[k0 + kk + 1];
        c = __builtin_amdgcn_wmma_f32_16x16x4_f32(false, a, false, b,
                                                  (short)0, c, false, false);
    }
    int rbase = m0 + ((lane >> 4) << 3);
#pragma unroll
    for (int r = 0; r < 8; ++r)
        AH[(size_t)(rbase + r) * N_NODES + n0 + mm] = c[r];
}

// ---------------- aggregation kernels ----------------

__global__ void k_init_agg(const float* __restrict__ H, const float* __restrict__ deginv,
                           float* __restrict__ agg, int Fshift) {
    int i = blockIdx.x * blockDim.x + threadIdx.x;
    agg[i] = H[i] * deginv[i >> Fshift];     // self-loop term, overwrites scratch
}

__global__ void k_scatter(const int* __restrict__ src, const int* __restrict__ dst,
                          const float* __restrict__ norm, const float* __restrict__ H,
                          float* __restrict__ agg, int Fshift) {
    int t = blockIdx.x * blockDim.x + threadIdx.x;
    int e = t >> Fshift;
    int f = t & ((1 << Fshift) - 1);
    float v = H[((size_t)src[e] << Fshift) + f] * norm[e];
    atomicAdd(&agg[((size_t)dst[e] << Fshift) + f], v);
}

__global__ void k_epilogue(const float* __restrict__ agg, const float* __restrict__ bias,
                           float* __restrict__ out, int Fshift, int relu) {
    int i = blockIdx.x * blockDim.x + threadIdx.x;
    int f = i & ((1 << Fshift) - 1);
    float v = agg[i] + bias[f];
    out[i] = relu ? fmaxf(v, 0.0f) : v;
}

// ---------------- host orchestration ----------------

static void gcn_layer(const float* x, const float* W, const float* b, float* out,
                      float* gbuf, float* agg,
                      const int* src, const int* dst, const float* norm,
                      const float* deginv, int Fin, int Fout, int Fshift, int relu,
                      hipStream_t stream) {
    int tiles = (N_NODES / 16) * (Fout / 16);
    k_gemm_xw<<<tiles / 8, 256, 0, stream>>>(x, W, gbuf, Fin, Fout);
    int nf = N_NODES * Fout;
    k_init_agg<<<nf / 256, 256, 0, stream>>>(gbuf, deginv, agg, Fshift);
    long ef = (long)N_EDGES * Fout;
    k_scatter<<<(int)(ef / 256), 256, 0, stream>>>(src, dst, norm, gbuf, agg, Fshift);
    k_epilogue<<<nf / 256, 256, 0, stream>>>(agg, b, out, Fshift, relu);
}

extern "C" void kernel_launch(void* const* d_in, const int* in_sizes, int n_in,
                              void* d_out, int out_size, void* d_ws, size_t ws_size,
                              hipStream_t stream) {
    const float* x   = (const float*)d_in[0];
    const int*   ei  = (const int*)d_in[1];
    const float* We1 = (const float*)d_in[2];
    const float* be1 = (const float*)d_in[3];
    const float* We2 = (const float*)d_in[4];
    const float* be2 = (const float*)d_in[5];
    const float* Wa1 = (const float*)d_in[6];
    const float* ba1 = (const float*)d_in[7];
    const float* Wa2 = (const float*)d_in[8];
    const float* ba2 = (const float*)d_in[9];
    const float* Ws  = (const float*)d_in[10];
    const float* bs  = (const float*)d_in[11];

    const int* src = ei;             // edge_index[0, :]
    const int* dst = ei + N_EDGES;   // edge_index[1, :]

    float* w = (float*)d_ws;
    float* deg    = w;  w += N_NODES;
    float* dinvs  = w;  w += N_NODES;
    float* deginv = w;  w += N_NODES;
    float* norm   = w;  w += N_EDGES;
    float* gbuf   = w;  w += (size_t)N_NODES * 256;   // pre-aggregation h = x@W
    float* agg    = w;  w += (size_t)N_NODES * 256;   // aggregation accumulator
    float* h1     = w;  w += (size_t)N_NODES * 128;   // encoder layer 1 output
    float* adec   = w;  w += (size_t)N_NODES * 128;   // attr decoder hidden
    float* sbuf   = w;  w += (size_t)N_NODES * 64;    // structure decoder output

    float* xhat = (float*)d_out;
    float* ahat = xhat + (size_t)N_NODES * 256;
    float* zout = ahat + (size_t)N_NODES * N_NODES;

    // degree / norm precompute
    k_init_deg  <<<N_NODES / 256, 256, 0, stream>>>(deg);
    k_count_deg <<<N_EDGES / 256, 256, 0, stream>>>(dst, deg);
    k_finish_deg<<<N_NODES / 256, 256, 0, stream>>>(deg, dinvs, deginv);
    k_norm      <<<N_EDGES / 256, 256, 0, stream>>>(src, dst, dinvs, norm);

    // encoder
    gcn_layer(x,    We1, be1, h1,   gbuf, agg, src, dst, norm, deginv, 256, 128, 7, 1, stream);
    gcn_layer(h1,   We2, be2, zout, gbuf, agg, src, dst, norm, deginv, 128,  64, 6, 1, stream);
    // attribute decoder
    gcn_layer(zout, Wa1, ba1, adec, gbuf, agg, src, dst, norm, deginv,  64, 128, 7, 1, stream);
    gcn_layer(adec, Wa2, ba2, xhat, gbuf, agg, src, dst, norm, deginv, 128, 256, 8, 0, stream);
    // structure decoder
    gcn_layer(zout, Ws,  bs,  sbuf, gbuf, agg, src, dst, norm, deginv,  64,  64, 6, 1, stream);

    // a_hat = s @ s^T  (store-bandwidth bound; L2-resident reads)
    dim3 grid_sst(N_NODES / 16 / 8, N_NODES / 16);
    k_gemm_sst<<<grid_sst, 256, 0, stream>>>(sbuf, ahat);
}